// BERTEmbedder_43550968381905
// MI455X (gfx1250) — compile-verified
//
#include <hip/hip_runtime.h>
#include <hip/hip_bf16.h>
#include <stdint.h>

// Problem constants (match reference)
#define B_ 16
#define S_ 4096
#define D_ 768
#define T_ 2048

#define NCHUNK 256                 // chunks per sequence
#define CL (S_ / NCHUNK)           // 16 rows per block
#define ROW_BYTES (D_ * 4)         // 3072 B per row
#define THREADS (D_ / 4)           // 192 threads, float4 per lane

// ---------------- CDNA5 async global->LDS helpers ----------------
__device__ __forceinline__ void async_row_load(uint32_t lds_addr, const float* gptr) {
#if defined(__gfx1250__)
  // One B128 transfer per active lane: copies 16 bytes from the global address in gptr to the LDS address in lds_addr, tracked by ASYNCcnt.
  asm volatile("global_load_async_to_lds_b128 %0, %1, off"
               :: "v"(lds_addr), "v"(gptr)
               : "memory");
#else
  (void)lds_addr; (void)gptr;
#endif
}

__device__ __forceinline__ void wait_async_le1() {
#if defined(__gfx1250__)
#if __has_builtin(__builtin_amdgcn_s_wait_asynccnt)
  __builtin_amdgcn_s_wait_asynccnt(1);
#else
  asm volatile("s_wait_asynccnt 0x1" ::: "memory");
#endif
#endif
}

__device__ __forceinline__ void wait_async_le0() {
#if defined(__gfx1250__)
#if __has_builtin(__builtin_amdgcn_s_wait_asynccnt)
  __builtin_amdgcn_s_wait_asynccnt(0);
#else
  asm volatile("s_wait_asynccnt 0x0" ::: "memory");
#endif
#endif
}

// ---------------- Pass 1: zero output + counts ----------------
__global__ void zero_kernel(float4* __restrict__ out4, float4* __restrict__ cnt4,
                            int n_out4, int n_cnt4) {
  int i = blockIdx.x * blockDim.x + threadIdx.x;
  const float4 z = make_float4(0.f, 0.f, 0.f, 0.f);
  if (i < n_out4) {
    out4[i] = z;
  } else {
    int j = i - n_out4;
    if (j < n_cnt4) cnt4[j] = z;
  }
}

// ---------------- Pass 2: segment-sum with async LDS double buffering ----------------
__global__ __launch_bounds__(THREADS)
void seg_accum_kernel(const float* __restrict__ x, const int* __restrict__ seg,
                      float* __restrict__ out, float* __restrict__ cnts) {
  __shared__ __align__(16) float buf[2][D_];   // 2 x 3KB double buffer

  const int tid   = threadIdx.x;
  const int blk   = blockIdx.x;
  const int b     = blk / NCHUNK;
  const int chunk = blk % NCHUNK;
  const int r0    = chunk * CL;

  const float* xrow0 = x + ((size_t)b * S_ + r0) * D_;
  // Low 32 bits of a generic pointer to a shared object are the LDS byte offset
  // (ISA 10.2 aperture mapping): use as VDST for the async load.
  const uint32_t lds0 = (uint32_t)(uintptr_t)(&buf[0][0]) + (uint32_t)(tid * 16);

  // Prefetch row 0 into buffer 0.
  async_row_load(lds0, xrow0 + tid * 4);

  const int* segrow = seg + (size_t)b * S_ + r0;
  int    cur   = segrow[0];
  float4 acc   = make_float4(0.f, 0.f, 0.f, 0.f);
  float  cnt   = 0.f;
  bool   first = true;                         // first segment of chunk may span backwards

  for (int i = 0; i < CL; ++i) {
    if (i + 1 < CL) {
      // Prefetch next row into the other buffer, then wait for the current row
      // (async loads complete in order, ASYNCcnt is per-wave; each lane reads
      // back exactly the 16 bytes it requested, so no block barrier is needed).
      async_row_load(lds0 + (uint32_t)(((i + 1) & 1) * ROW_BYTES),
                     xrow0 + (size_t)(i + 1) * D_ + tid * 4);
      wait_async_le1();
    } else {
      wait_async_le0();
    }

    const int sid = segrow[i];
    if (sid != cur) {
      float* op = out + ((size_t)b * T_ + cur) * D_ + tid * 4;
      if (first) {
        // Segment may have started in the previous chunk -> atomic merge.
        atomicAdd(op + 0, acc.x);
        atomicAdd(op + 1, acc.y);
        atomicAdd(op + 2, acc.z);
        atomicAdd(op + 3, acc.w);
        if (tid == 0) atomicAdd(cnts + (size_t)b * T_ + cur, cnt);
      } else {
        // Interior segment: sole writer, plain vector store over zeros.
        *(float4*)op = acc;
        if (tid == 0) cnts[(size_t)b * T_ + cur] = cnt;
      }
      acc = make_float4(0.f, 0.f, 0.f, 0.f);
      cnt = 0.f;
      cur = sid;
      first = false;
    }

    const float4 v = *(const float4*)(&buf[i & 1][tid * 4]);
    acc.x += v.x; acc.y += v.y; acc.z += v.z; acc.w += v.w;
    cnt += 1.f;
  }

  // Final segment may continue into the next chunk -> always atomic.
  float* op = out + ((size_t)b * T_ + cur) * D_ + tid * 4;
  atomicAdd(op + 0, acc.x);
  atomicAdd(op + 1, acc.y);
  atomicAdd(op + 2, acc.z);
  atomicAdd(op + 3, acc.w);
  if (tid == 0) atomicAdd(cnts + (size_t)b * T_ + cur, cnt);
}

// ---------------- Pass 3: divide by max(count, 1) ----------------
__global__ void finalize_kernel(float* __restrict__ out, const float* __restrict__ cnts) {
  const int n4 = (B_ * T_ * D_) / 4;
  int i4 = blockIdx.x * blockDim.x + threadIdx.x;
  if (i4 >= n4) return;
  const int row = i4 / (D_ / 4);               // which (b, t) this float4 belongs to
  const float c   = cnts[row];
  const float inv = 1.0f / fmaxf(c, 1.0f);
  float4 v = ((const float4*)out)[i4];
  v.x *= inv; v.y *= inv; v.z *= inv; v.w *= inv;
  ((float4*)out)[i4] = v;
}

extern "C" void kernel_launch(void* const* d_in, const int* in_sizes, int n_in,
                              void* d_out, int out_size, void* d_ws, size_t ws_size,
                              hipStream_t stream) {
  (void)in_sizes; (void)n_in; (void)out_size; (void)ws_size;

  const float* x   = (const float*)d_in[0];   // [B, S, D] fp32
  const int*   seg = (const int*)  d_in[1];   // [B, S] int32, sorted per row
  float*       out = (float*)d_out;           // [B, T, D] fp32
  float*       cnt = (float*)d_ws;            // [B, T] fp32 counts (128 KB scratch)

  const int n_out4 = (B_ * T_ * D_) / 4;      // 6,291,456
  const int n_cnt4 = (B_ * T_) / 4;           // 8,192
  const int totz   = n_out4 + n_cnt4;

  zero_kernel<<<(totz + 255) / 256, 256, 0, stream>>>(
      (float4*)out, (float4*)cnt, n_out4, n_cnt4);

  seg_accum_kernel<<<B_ * NCHUNK, THREADS, 0, stream>>>(x, seg, out, cnt);

  finalize_kernel<<<(n_out4 + 255) / 256, 256, 0, stream>>>(out, cnt);
}